// IntegrateWithWeights3DModule_89790586290717
// MI455X (gfx1250) — compile-verified
//
#include <hip/hip_runtime.h>
#include <climits>
#include <cstdint>

// ---------------------------------------------------------------------------
// Ragged per-ray weighted integration (segment-sum with sorted keys).
// Memory-bound streaming kernel: CDNA5 async global->LDS double buffering
// (GLOBAL_LOAD_ASYNC_TO_LDS_B128 / S_WAIT_ASYNCCNT), wave32 segmented scan
// with carry injection at lane 0, boundary atomics.
// out[r] = sum_i [idx_i==r] * w_i * v_i.
// ---------------------------------------------------------------------------

#define WPB    8      // waves per block (8 * 32 = 256 threads)
#define SUPER  128    // samples per super-tile staged in LDS

__device__ __forceinline__ void issue_async_b128(uint32_t lds_addr, uint64_t gaddr) {
  // CDNA5 async copy: 16 bytes per lane, global -> LDS, tracked by ASYNCcnt.
  asm volatile("global_load_async_to_lds_b128 %0, %1, off"
               :: "v"(lds_addr), "v"(gaddr)
               : "memory");
}

__global__ __launch_bounds__(WPB * 32)
void integrate_zero_out_kernel(float* __restrict__ out, int n) {
  int i = blockIdx.x * blockDim.x + threadIdx.x;
  if (i < n) out[i] = 0.0f;
}

__global__ __launch_bounds__(WPB * 32)
void integrate_seg_sum_kernel(const int*   __restrict__ ridx,
                              const float* __restrict__ vals,
                              const float* __restrict__ wts,
                              float*       __restrict__ out,
                              long long Ntot, long long spw) {
  __shared__ __align__(16) float sv[WPB][2][SUPER * 3];
  __shared__ __align__(16) float sw[WPB][2][SUPER];
  __shared__ __align__(16) int   si[WPB][2][SUPER];

  const int lane = threadIdx.x & 31;
  const int wid  = threadIdx.x >> 5;

  const long long gw         = (long long)blockIdx.x * WPB + wid;
  const long long chunkStart = gw * spw;
  if (chunkStart >= Ntot) return;
  long long chunkEnd = chunkStart + spw;
  if (chunkEnd > Ntot) chunkEnd = Ntot;
  const int numSuper = (int)((chunkEnd - chunkStart + (SUPER - 1)) / SUPER);

  float* svp[2] = { &sv[wid][0][0], &sv[wid][1][0] };
  float* swp[2] = { &sw[wid][0][0], &sw[wid][1][0] };
  int*   sip[2] = { &si[wid][0][0], &si[wid][1][0] };
  uint32_t ldsV[2], ldsW[2], ldsI[2];
  #pragma unroll
  for (int b = 0; b < 2; ++b) {
    // Low 32 bits of a flat LDS pointer == workgroup-relative LDS byte address
    // (flat LDS aperture truncation; async engine adds LDS_BASE itself).
    ldsV[b] = (uint32_t)(uintptr_t)svp[b];
    ldsW[b] = (uint32_t)(uintptr_t)swp[b];
    ldsI[b] = (uint32_t)(uintptr_t)sip[b];
  }

  const uint64_t vG = (uint64_t)(uintptr_t)vals;
  const uint64_t wG = (uint64_t)(uintptr_t)wts;
  const uint64_t iG = (uint64_t)(uintptr_t)ridx;
  const uint32_t lb = (uint32_t)lane * 16u;

  // Fill one LDS super-tile buffer; returns true if async copies were issued.
  auto fill = [&](int buf, long long sb) -> bool {
    if (sb + SUPER <= Ntot) {
      // Full tile: 5 x b128 per lane covers 1536B values + 512B weights + 512B idx.
      const uint64_t v0 = vG + (uint64_t)sb * 12u + lb;
      issue_async_b128(ldsV[buf] + lb,         v0);
      issue_async_b128(ldsV[buf] + lb + 512u,  v0 + 512u);
      issue_async_b128(ldsV[buf] + lb + 1024u, v0 + 1024u);
      issue_async_b128(ldsW[buf] + lb, wG + (uint64_t)sb * 4u + lb);
      issue_async_b128(ldsI[buf] + lb, iG + (uint64_t)sb * 4u + lb);
      return true;
    }
    // Tail tile (at most one wave hits this): direct bounds-checked fill.
    for (int j = lane; j < SUPER; j += 32) {
      const long long s = sb + j;
      if (s < Ntot) {
        sip[buf][j]         = ridx[s];
        swp[buf][j]         = wts[s];
        svp[buf][3 * j + 0] = vals[3 * s + 0];
        svp[buf][3 * j + 1] = vals[3 * s + 1];
        svp[buf][3 * j + 2] = vals[3 * s + 2];
      }
    }
    return false;
  };

  // Running segment carry (held uniformly across the wave).
  int   cidx = INT_MAX;
  float cx = 0.f, cy = 0.f, cz = 0.f;

  bool curAsync = fill(0, chunkStart);
  for (int st = 0; st < numSuper; ++st) {
    const int       buf = st & 1;
    const long long sb  = chunkStart + (long long)st * SUPER;

    bool nextAsync = false;
    if (st + 1 < numSuper) nextAsync = fill(buf ^ 1, sb + SUPER);

    if (curAsync) {
      if (nextAsync) {
        // Current tile's 5 async loads retire in order; keep next 5 in flight.
        asm volatile("s_wait_asynccnt 0x5" ::: "memory");
      } else {
        asm volatile("s_wait_asynccnt 0x0" ::: "memory");
      }
    }

    #pragma unroll
    for (int t = 0; t < SUPER / 32; ++t) {
      const long long tb = sb + (long long)t * 32;
      if (tb >= chunkEnd) break;

      const int       sl = t * 32 + lane;
      const long long s  = sb + sl;

      int   key = INT_MAX;
      float x = 0.f, y = 0.f, z = 0.f;
      if (s < chunkEnd) {
        key = sip[buf][sl];
        const float ww = swp[buf][sl];
        x = svp[buf][3 * sl + 0] * ww;
        y = svp[buf][3 * sl + 1] * ww;
        z = svp[buf][3 * sl + 2] * ww;
      }

      // Carry handling BEFORE the scan (lane 0 always holds the tile's first
      // key, and tiles always contain at least one valid sample):
      //  - if the first segment continues the carried segment, inject the
      //    carry into lane 0 so the scan propagates it through the segment;
      //  - otherwise the carried segment ended exactly at the previous tile
      //    boundary -> lane 0 flushes the stale carry now.
      if (lane == 0 && cidx != INT_MAX) {
        if (key == cidx) {
          x += cx; y += cy; z += cz;
        } else {
          atomicAdd(&out[3 * cidx + 0], cx);
          atomicAdd(&out[3 * cidx + 1], cy);
          atomicAdd(&out[3 * cidx + 2], cz);
        }
      }

      // Key-compare segmented inclusive scan across the wave32.
      #pragma unroll
      for (int d = 1; d < 32; d <<= 1) {
        const int   ok = __shfl_up(key, d);
        const float ox = __shfl_up(x, d);
        const float oy = __shfl_up(y, d);
        const float oz = __shfl_up(z, d);
        if (lane >= d && ok == key) { x += ox; y += oy; z += oz; }
      }

      // Segment tails (not lane 31) flush their full in-tile segment sum
      // (carry, if any, was already folded in by the lane-0 injection).
      const int nk = __shfl_down(key, 1);
      if (lane < 31 && key != nk && key != INT_MAX) {
        atomicAdd(&out[3 * key + 0], x);
        atomicAdd(&out[3 * key + 1], y);
        atomicAdd(&out[3 * key + 2], z);
      }

      // Lane 31's running segment becomes the new carry. If lane 31 is an
      // invalid (masked) lane the carry becomes the INT_MAX sentinel with a
      // zero value, which is exactly the "no carry" state.
      cidx = __shfl(key, 31);
      cx   = __shfl(x, 31);
      cy   = __shfl(y, 31);
      cz   = __shfl(z, 31);
    }

    curAsync = nextAsync;
  }

  if (lane == 0 && cidx != INT_MAX) {
    atomicAdd(&out[3 * cidx + 0], cx);
    atomicAdd(&out[3 * cidx + 1], cy);
    atomicAdd(&out[3 * cidx + 2], cz);
  }
}

extern "C" void kernel_launch(void* const* d_in, const int* in_sizes, int n_in,
                              void* d_out, int out_size, void* d_ws, size_t ws_size,
                              hipStream_t stream) {
  (void)n_in; (void)d_ws; (void)ws_size;
  const int*   ridx = (const int*)d_in[0];     // [N] sorted ray indices
  const float* vals = (const float*)d_in[1];   // [N,3]
  const float* wts  = (const float*)d_in[2];   // [N,1]
  // d_in[3] = n_rays scalar (device) — out_size already encodes n_rays*3.
  float* out = (float*)d_out;

  const long long N = (long long)in_sizes[0];

  // Zero the accumulator output (harness poisons d_out before timing).
  {
    const int zb = (out_size + 255) / 256;
    if (zb > 0)
      integrate_zero_out_kernel<<<zb, 256, 0, stream>>>(out, out_size);
  }
  if (N <= 0) return;

  const int blocks = 1024;                       // 8192 wave32 chunks
  const long long totalWaves = (long long)blocks * WPB;
  long long spw = (N + totalWaves - 1) / totalWaves;
  spw = ((spw + SUPER - 1) / SUPER) * SUPER;     // super-tile aligned chunks

  integrate_seg_sum_kernel<<<blocks, WPB * 32, 0, stream>>>(
      ridx, vals, wts, out, N, spw);
}